// CodeBook_83683142795648
// MI455X (gfx1250) — compile-verified
//
#include <hip/hip_runtime.h>

// VQ codebook quantization for MI455X (gfx1250, wave32).
// Sizes fixed by the reference: B=64, C=256, H=W=32, D=128, K=1024.
//   N_ROWS = B*H*W = 65536
// Workspace layout (d_ws): zp[N_ROWS*128] f32 (32MB) | hn[1024] f32 | idx[N_ROWS] i32
// Output layout (d_out):   rep_z_q [64,128,32,32] f32 (8388608) | min_indices as f32 [65536]

typedef __attribute__((ext_vector_type(2))) float v2f;
typedef __attribute__((ext_vector_type(4))) float v4f;
typedef __attribute__((ext_vector_type(8))) float v8f;

#define HW_SZ   1024
#define C_DIM   256
#define D_DIM   128
#define K_CODES 1024
#define N_ROWS  65536

__device__ __forceinline__ v8f wmma4(v2f a, v2f b, v8f c) {
  // V_WMMA_F32_16X16X4_F32: D = A(16x4) * B(4x16) + C(16x16), all fp32
  return __builtin_amdgcn_wmma_f32_16x16x4_f32(
      /*neg_a=*/false, a, /*neg_b=*/false, b,
      /*c_mod=*/(short)0, c, /*reuse_a=*/false, /*reuse_b=*/false);
}

// ---------------------------------------------------------------------------
// Kernel 0: half squared norms of codebook rows: hn[k] = 0.5 * sum_d e[k,d]^2
// ---------------------------------------------------------------------------
__global__ void __launch_bounds__(256) vq_norms_kernel(
    const float* __restrict__ embed, float* __restrict__ hn) {
  int k = blockIdx.x * 256 + threadIdx.x;
  if (k < K_CODES) {
    const float* e = embed + (size_t)k * D_DIM;
    float s = 0.f;
    #pragma unroll 4
    for (int d = 0; d < D_DIM; ++d) { float v = e[d]; s += v * v; }
    hn[k] = 0.5f * s;
  }
}

// ---------------------------------------------------------------------------
// Kernel 1: projection GEMM  zp[n,d] = sum_c z[b,c,hw] * pw[d,c] + pb[d]
// M=65536 (n = b*1024+hw), N=128, K=256. One wave computes a 16x128 strip.
// ---------------------------------------------------------------------------
__global__ void __launch_bounds__(256) vq_proj_kernel(
    const float* __restrict__ z, const float* __restrict__ pw,
    const float* __restrict__ pb, float* __restrict__ zp) {
  const int lane  = threadIdx.x & 31;
  const int wid   = threadIdx.x >> 5;
  const int mtile = blockIdx.x * 8 + wid;        // 4096 tiles total
  const int row0  = mtile * 16;
  const int b     = row0 >> 10;                  // 16 | 1024, same batch per tile
  const int hw0   = row0 & 1023;
  const int m     = lane & 15;                   // A: M index
  const int kh    = lane >> 4;                   // A/B: K half (K = k0 + 2*kh + j)

  const float* zbase = z + (size_t)b * (C_DIM * HW_SZ) + hw0 + m;

  v8f acc[8];
  #pragma unroll
  for (int nt = 0; nt < 8; ++nt)
    acc[nt] = (v8f){0.f,0.f,0.f,0.f,0.f,0.f,0.f,0.f};

  #pragma unroll 1
  for (int k0 = 0; k0 < C_DIM; k0 += 4) {
    const int ka = k0 + 2 * kh;
    v2f a;
    a.x = zbase[(size_t)(ka + 0) * HW_SZ];
    a.y = zbase[(size_t)(ka + 1) * HW_SZ];
    if (k0 + 16 < C_DIM)                          // stream-ahead on the 64MB z read
      __builtin_prefetch(zbase + (size_t)(k0 + 16 + 2 * kh) * HW_SZ, 0, 1);
    #pragma unroll
    for (int nt = 0; nt < 8; ++nt) {
      // B[k, n] = pw[n*C + k], n = nt*16 + m  (even offset -> 8B-aligned b64 load)
      v2f bf = *(const v2f*)(pw + (size_t)(nt * 16 + m) * C_DIM + ka);
      acc[nt] = wmma4(a, bf, acc[nt]);
    }
  }

  // C/D layout: vgpr r, lanes 0-15 -> M=r ; lanes 16-31 -> M=r+8 ; N = lane&15
  const int nlo = lane & 15;
  const int mh  = (lane >> 4) * 8;
  #pragma unroll
  for (int nt = 0; nt < 8; ++nt) {
    float bias = pb[nt * 16 + nlo];
    #pragma unroll
    for (int r = 0; r < 8; ++r) {
      int row = row0 + mh + r;
      zp[(size_t)row * D_DIM + nt * 16 + nlo] = acc[nt][r] + bias;
    }
  }
}

// ---------------------------------------------------------------------------
// Kernel 2: fused score GEMM + argmin.
//   score[n,k] = zp[n,:]·e[k,:] - 0.5*||e_k||^2 ;  idx[n] = argmax_k score
// (equivalent to argmin of squared distance; ||z||^2 row-constant dropped)
// Block = 256 threads = 8 waves; each wave owns a 16-row strip whose A
// fragments live entirely in registers (32 x v2f, invariant across the 64
// code tiles); the block shares each staged 16-code B tile via LDS.
// Two interleaved accumulators keep consecutive WMMAs independent.
// ---------------------------------------------------------------------------
__global__ void __launch_bounds__(256) vq_argmin_kernel(
    const float* __restrict__ zp, const float* __restrict__ embed,
    const float* __restrict__ hn, int* __restrict__ idx_out,
    float* __restrict__ idxf_out) {
  __shared__ float ldsB[16 * D_DIM];       // 8 KB: 16 codes x 128

  const int tid   = threadIdx.x;
  const int lane  = tid & 31;
  const int wid   = tid >> 5;
  const int mtile = blockIdx.x * 8 + wid;
  const int row0  = mtile * 16;
  const int nlo   = lane & 15;
  const int kh    = lane >> 4;

  // Preload this wave's A strip into registers: frag i covers K = 4i..4i+3;
  // this lane holds (M=nlo, K=4i+2*kh, 4i+2*kh+1). zp is L2-resident.
  v2f afrag[32];
  {
    const float* arow = zp + (size_t)(row0 + nlo) * D_DIM + 2 * kh;
    #pragma unroll
    for (int i = 0; i < 32; ++i)
      afrag[i] = *(const v2f*)(arow + i * 4);
  }

  const float* Bbase = ldsB + nlo * D_DIM + 2 * kh;   // code N = nlo

  float best[8];
  int   bidx[8];
  #pragma unroll
  for (int r = 0; r < 8; ++r) { best[r] = -3.4e38f; bidx[r] = 0; }

  for (int nt = 0; nt < 64; ++nt) {
    __syncthreads();                      // previous tile's readers done
    {                                     // stage B tile: codes nt*16..nt*16+15
      const v4f* src = (const v4f*)(embed + (size_t)nt * 16 * D_DIM);
      v4f* dst = (v4f*)ldsB;              // 512 x v4f, 2 per thread (b128 copies)
      dst[tid]       = src[tid];
      dst[tid + 256] = src[tid + 256];
    }
    __syncthreads();

    v8f acc0 = (v8f){0.f,0.f,0.f,0.f,0.f,0.f,0.f,0.f};
    v8f acc1 = (v8f){0.f,0.f,0.f,0.f,0.f,0.f,0.f,0.f};
    #pragma unroll
    for (int i = 0; i < 32; i += 2) {
      v2f b0 = *(const v2f*)(Bbase + 4 * i);        // ds_load_b64
      v2f b1 = *(const v2f*)(Bbase + 4 * i + 4);
      acc0 = wmma4(afrag[i],     b0, acc0);
      acc1 = wmma4(afrag[i + 1], b1, acc1);
    }

    const float h = hn[nt * 16 + nlo];
    const int   n = nt * 16 + nlo;          // this lane's column (code id)
    #pragma unroll
    for (int r = 0; r < 8; ++r) {
      float s = (acc0[r] + acc1[r]) - h;
      if (s > best[r]) { best[r] = s; bidx[r] = n; }  // strict '>' keeps lowest n
    }
  }

  // Reduce across the 16 lanes of each half-wave (columns n%16 classes),
  // lowest index wins ties (matches jnp.argmin semantics).
  #pragma unroll
  for (int r = 0; r < 8; ++r) {
    float bv = best[r]; int bi = bidx[r];
    #pragma unroll
    for (int off = 8; off >= 1; off >>= 1) {
      float ov = __shfl_xor(bv, off, 16);
      int   oi = __shfl_xor(bi, off, 16);
      if (ov > bv || (ov == bv && oi < bi)) { bv = ov; bi = oi; }
    }
    if (nlo == 0) {
      int row = row0 + kh * 8 + r;
      idx_out[row]  = bi;
      idxf_out[row] = (float)bi;
    }
  }
}

// ---------------------------------------------------------------------------
// Kernel 3: gather + NHWC->NCHW transpose via LDS.
//   out[b,d,h,w] = embed[idx[b*1024+hw], d]
// Block handles 64 consecutive rows n (same batch b); padded LDS (129) avoids
// bank conflicts on the transposed read.
// ---------------------------------------------------------------------------
__global__ void __launch_bounds__(256) vq_gather_kernel(
    const float* __restrict__ embed, const int* __restrict__ idx,
    float* __restrict__ out) {
  __shared__ float q[64 * 129];            // ~33 KB
  const int tid = threadIdx.x;
  const int n0  = blockIdx.x * 64;
  const int b   = n0 >> 10;
  const int hw0 = n0 & 1023;

  #pragma unroll 4
  for (int i = 0; i < 32; ++i) {
    int flat = tid + i * 256;              // 0..8191
    int r = flat >> 7;                     // local row 0..63
    int d = flat & 127;
    q[r * 129 + d] = embed[(size_t)idx[n0 + r] * D_DIM + d];
  }
  __syncthreads();
  #pragma unroll 4
  for (int i = 0; i < 32; ++i) {
    int flat  = tid + i * 256;
    int d     = flat >> 6;                 // 0..127
    int hwoff = flat & 63;
    out[((size_t)(b * D_DIM + d) << 10) + hw0 + hwoff] = q[hwoff * 129 + d];
  }
}

// ---------------------------------------------------------------------------
extern "C" void kernel_launch(void* const* d_in, const int* in_sizes, int n_in,
                              void* d_out, int out_size, void* d_ws, size_t ws_size,
                              hipStream_t stream) {
  const float* z     = (const float*)d_in[0];   // [64,256,32,32]
  const float* pw    = (const float*)d_in[1];   // [128,256]
  const float* pb    = (const float*)d_in[2];   // [128]
  const float* embed = (const float*)d_in[3];   // [1024,128]

  float* out  = (float*)d_out;                          // rep_z_q, 8388608 floats
  float* idxf = out + (size_t)N_ROWS * D_DIM;           // indices (as float) tail

  float* zp  = (float*)d_ws;                            // 32 MB
  float* hn  = zp + (size_t)N_ROWS * D_DIM;             // 4 KB
  int*   idi = (int*)(hn + K_CODES);                    // 256 KB

  hipLaunchKernelGGL(vq_norms_kernel,  dim3(K_CODES / 256), dim3(256), 0, stream,
                     embed, hn);
  hipLaunchKernelGGL(vq_proj_kernel,   dim3(512),  dim3(256), 0, stream,
                     z, pw, pb, zp);
  hipLaunchKernelGGL(vq_argmin_kernel, dim3(512),  dim3(256), 0, stream,
                     zp, embed, hn, idi, idxf);
  hipLaunchKernelGGL(vq_gather_kernel, dim3(1024), dim3(256), 0, stream,
                     embed, idi, out);
}